// yoloLoss_40226663694840
// MI455X (gfx1250) — compile-verified
//
#include <hip/hip_runtime.h>
#include <hip/hip_bf16.h>

// ---------------------------------------------------------------------------
// YOLO loss on MI455X (gfx1250, wave32).
// Bandwidth-bound: 96 MB streamed once -> ~4.1 us floor at 23.3 TB/s.
// Strategy: 1 thread per (b,i,j) cell, b64 nontemporal vector loads,
// all-register per-cell math, WMMA-based wave reduction, deterministic
// two-pass block reduction through d_ws.
// ---------------------------------------------------------------------------

typedef __attribute__((ext_vector_type(2))) float f32x2;
typedef __attribute__((ext_vector_type(8))) float f32x8;

#define CELL_C 30
#define GRID_S 14

// Wave32 sum using V_WMMA_F32_16X16X4_F32.
// A layout (32-bit A, 16x4): lane L<16 holds A[L][K=0] in VGPR0, A[L][K=1] in
// VGPR1; lane L>=16 holds A[L-16][K=2], A[L-16][K=3]. With a = {v, 0} and
// B = all-ones, D[i][j] = v_i + v_{i+16} (replicated over j). Each lane sums
// its 8 accumulator regs (rows 0-7 on lanes<16, rows 8-15 on lanes>=16);
// shfl_xor(16) combines the two halves -> full wave sum on every lane.
// Requires EXEC all-1s (call only from re-converged code).
__device__ __forceinline__ float wave_sum32(float v) {
#if defined(__gfx1250__) && __has_builtin(__builtin_amdgcn_wmma_f32_16x16x4_f32)
  f32x2 a;
  a.x = v;
  a.y = 0.0f;
  f32x2 b;
  b.x = 1.0f;
  b.y = 1.0f;
  f32x8 c = {};
  c = __builtin_amdgcn_wmma_f32_16x16x4_f32(
      /*neg_a=*/false, a, /*neg_b=*/false, b,
      /*c_mod=*/(short)0, c, /*reuse_a=*/false, /*reuse_b=*/false);
  float s = ((c[0] + c[1]) + (c[2] + c[3])) + ((c[4] + c[5]) + (c[6] + c[7]));
  s += __shfl_xor(s, 16, 32);
  return s;
#else
  for (int off = 16; off > 0; off >>= 1) v += __shfl_xor(v, off, 32);
  return v;
#endif
}

// Per-cell YOLO loss; pv/tv are the 30 channel values held in registers
// (all indexing below is compile-time constant after unrolling).
__device__ __forceinline__ float cell_loss(const float* pv, const float* tv) {
  const float inv14 = 1.0f / 14.0f;

  const float coo = (tv[4] > 0.0f) ? 1.0f : 0.0f;
  const float noo = 1.0f - coo;

  // no-object confidence loss (channels 4 and 9)
  const float d4 = pv[4] - tv[4];
  const float d9 = pv[9] - tv[9];
  const float noo_loss = noo * (d4 * d4 + d9 * d9);

  // class loss (channels 10..29)
  float cls = 0.0f;
#pragma unroll
  for (int k = 10; k < CELL_C; ++k) {
    const float d = pv[k] - tv[k];
    cls = fmaf(d, d, cls);
  }

  // target box 0 in xyxy
  const float tcx = tv[0] * inv14, tcy = tv[1] * inv14;
  const float tw = tv[2], th = tv[3];
  const float tx1 = tcx - 0.5f * tw, ty1 = tcy - 0.5f * th;
  const float tx2 = tcx + 0.5f * tw, ty2 = tcy + 0.5f * th;
  const float ta = tw * th;

  // IoU of both pred boxes vs target box 0
  float iou0, iou1;
#pragma unroll
  for (int bb = 0; bb < 2; ++bb) {
    const float bx = pv[5 * bb + 0] * inv14;
    const float by = pv[5 * bb + 1] * inv14;
    const float bw = pv[5 * bb + 2];
    const float bh = pv[5 * bb + 3];
    const float x1 = bx - 0.5f * bw, y1 = by - 0.5f * bh;
    const float x2 = bx + 0.5f * bw, y2 = by + 0.5f * bh;
    float iw = fminf(x2, tx2) - fmaxf(x1, tx1);
    float ih = fminf(y2, ty2) - fmaxf(y1, ty1);
    iw = fmaxf(iw, 0.0f);
    ih = fmaxf(ih, 0.0f);
    const float inter = iw * ih;
    const float iou = inter / (bw * bh + ta - inter);
    if (bb == 0) iou0 = iou; else iou1 = iou;
  }

  // responsible box = argmax (first-occurrence tie-break like jnp.argmax)
  const bool sel1 = iou1 > iou0;
  const float max_iou = sel1 ? iou1 : iou0;

  const float prx = sel1 ? pv[5] : pv[0];
  const float pry = sel1 ? pv[6] : pv[1];
  const float prw = sel1 ? pv[7] : pv[2];
  const float prh = sel1 ? pv[8] : pv[3];
  const float prc = sel1 ? pv[9] : pv[4];
  const float trx = sel1 ? tv[5] : tv[0];
  const float try_ = sel1 ? tv[6] : tv[1];
  const float trw = sel1 ? tv[7] : tv[2];
  const float trh = sel1 ? tv[8] : tv[3];
  const float nrc = sel1 ? pv[4] : pv[9];  // non-responsible conf

  const float dconf = prc - max_iou;
  const float contain = dconf * dconf;

  const float dx = prx - trx;
  const float dy = pry - try_;
  const float dw = __builtin_sqrtf(prw) - __builtin_sqrtf(trw);
  const float dh = __builtin_sqrtf(prh) - __builtin_sqrtf(trh);
  const float loc = (dx * dx + dy * dy) + (dw * dw + dh * dh);

  const float not_contain = nrc * nrc;

  return coo * (5.0f * loc + 2.0f * contain + not_contain + cls) +
         0.5f * noo_loss;
}

__global__ void __launch_bounds__(256)
yolo_cell_kernel(const float* __restrict__ pred, const float* __restrict__ targ,
                 float* __restrict__ partials, int n_cells) {
  const int cell = blockIdx.x * blockDim.x + threadIdx.x;

  float loss = 0.0f;
  if (cell < n_cells) {
    // 30 floats = 120 B per cell, 8-byte aligned -> 15 x b64 NT loads each.
    const f32x2* p2 = reinterpret_cast<const f32x2*>(pred) + cell * 15;
    const f32x2* t2 = reinterpret_cast<const f32x2*>(targ) + cell * 15;
    float pv[CELL_C], tv[CELL_C];
#pragma unroll
    for (int i = 0; i < 15; ++i) {
      const f32x2 a = __builtin_nontemporal_load(p2 + i);
      const f32x2 b = __builtin_nontemporal_load(t2 + i);
      pv[2 * i] = a.x;
      pv[2 * i + 1] = a.y;
      tv[2 * i] = b.x;
      tv[2 * i + 1] = b.y;
    }
    loss = cell_loss(pv, tv);
  }

  // re-converged here: EXEC all-1s -> WMMA wave reduction is legal
  const float wsum = wave_sum32(loss);

  __shared__ float lds[8];
  const int lane = threadIdx.x & 31;
  const int wave = threadIdx.x >> 5;
  if (lane == 0) lds[wave] = wsum;
  __syncthreads();
  if (threadIdx.x == 0) {
    float s = 0.0f;
#pragma unroll
    for (int i = 0; i < 8; ++i) s += lds[i];
    partials[blockIdx.x] = s;
  }
}

__global__ void __launch_bounds__(256)
yolo_finalize_kernel(const float* __restrict__ partials, int n,
                     float* __restrict__ out, float inv_n_batch) {
  float s = 0.0f;
  for (int i = threadIdx.x; i < n; i += 256) s += partials[i];

  const float wsum = wave_sum32(s);

  __shared__ float lds[8];
  const int lane = threadIdx.x & 31;
  const int wave = threadIdx.x >> 5;
  if (lane == 0) lds[wave] = wsum;
  __syncthreads();
  if (threadIdx.x == 0) {
    float t = 0.0f;
#pragma unroll
    for (int i = 0; i < 8; ++i) t += lds[i];
    out[0] = t * inv_n_batch;
  }
}

extern "C" void kernel_launch(void* const* d_in, const int* in_sizes, int n_in,
                              void* d_out, int out_size, void* d_ws,
                              size_t ws_size, hipStream_t stream) {
  (void)n_in;
  (void)out_size;
  (void)ws_size;
  const float* pred = (const float*)d_in[0];
  const float* targ = (const float*)d_in[1];
  float* out = (float*)d_out;

  const int total = in_sizes[0];                 // B * 14 * 14 * 30
  const int n_cells = total / CELL_C;            // B * 196
  const int n_batch = n_cells / (GRID_S * GRID_S);  // B

  const int threads = 256;
  const int blocks = (n_cells + threads - 1) / threads;  // 1568 for B=2048

  float* partials = (float*)d_ws;  // blocks * 4 bytes of scratch

  yolo_cell_kernel<<<blocks, threads, 0, stream>>>(pred, targ, partials,
                                                   n_cells);
  yolo_finalize_kernel<<<1, 256, 0, stream>>>(partials, blocks, out,
                                              1.0f / (float)n_batch);
}